// TTTLayer_37374805409930
// MI455X (gfx1250) — compile-verified
//
#include <hip/hip_runtime.h>
#include <hip/hip_bf16.h>

#define HIDDEN 2048
#define HEADS  16
#define HEAD   128
#define SEQ    2048
#define BATCH  4
#define MBS    16
#define NCHUNK (SEQ / MBS)     // 128
#define ROWS   (BATCH * SEQ)   // 8192

#define KSLICE 16              // K-depth staged in LDS per GEMM stage
#define BPAD   20              // padded LDS row stride (floats): 16B-aligned + conflict-free

typedef float v2f __attribute__((ext_vector_type(2)));
typedef float v8f __attribute__((ext_vector_type(8)));

static __device__ __forceinline__ v8f vzero8() {
  v8f z = {0.f, 0.f, 0.f, 0.f, 0.f, 0.f, 0.f, 0.f};
  return z;
}

// D = A(16x4) * B(4x16) + C, full fp32 (the only fp32 WMMA on CDNA5).
static __device__ __forceinline__ v8f wmma_f32(v2f a, v2f b, v8f c) {
  return __builtin_amdgcn_wmma_f32_16x16x4_f32(false, a, false, b, (short)0, c,
                                               false, false);
}

// ---------------------------------------------------------------------------
// Big GEMM: C[M=8192, N=2048] = A[M,2048] @ W.T   (W is [N, 2048], Linear conv.)
// Block: 256 thr = 8 waves; block tile 128(M) x 128(N); wave tile 16(M) x 128(N).
// W K-slices staged in LDS (shared by all 8 waves, 8x vmem dedup), software-
// pipelined: stage s+1 global loads are in flight during stage s WMMAs.
// mode 0: scatter into [B, H, S, D] (k/v projection, RoPE applied later)
// mode 1: out[row, col] = acc + resid[row, col]   (final proj + residual)
// ---------------------------------------------------------------------------
__global__ __launch_bounds__(256) void gemm_xWT(const float* __restrict__ A,
                                                const float* __restrict__ W,
                                                float* __restrict__ out,
                                                const float* __restrict__ resid,
                                                int mode) {
  __shared__ float Blds[128 * BPAD];                 // 10 KB staged W slice

  const int tid  = threadIdx.x;
  const int lane = tid & 31;
  const int wave = tid >> 5;
  const int r    = lane & 15;       // M row within tile / N col within tile
  const int hi   = lane >> 4;       // lane-half selector
  const int koff = hi * 2;          // K sub-offset per A/B layout tables
  const int m0   = blockIdx.x * 128 + wave * 16;
  const int n0   = blockIdx.y * 128;

  v8f acc[8] = {vzero8(), vzero8(), vzero8(), vzero8(),
                vzero8(), vzero8(), vzero8(), vzero8()};

  const float* Arow = A + (size_t)(m0 + r) * HIDDEN + koff;

  // Staging assignment: 128 rows x 4 float4 = 512 float4 / 256 threads = 2 each
  const int f0   = tid * 2;
  const int row0 = f0 >> 2,       c0 = (f0 & 3) * 4;
  const int row1 = (f0 + 1) >> 2, c1 = ((f0 + 1) & 3) * 4;
  const float* Wst0 = W + (size_t)(n0 + row0) * HIDDEN + c0;
  const float* Wst1 = W + (size_t)(n0 + row1) * HIDDEN + c1;

  // ---- prologue: prefetch stage 0 into registers ----
  float4 st0 = *(const float4*)(Wst0);
  float4 st1 = *(const float4*)(Wst1);
  v2f a_cur[4], a_nxt[4];
#pragma unroll
  for (int u = 0; u < 4; ++u) a_cur[u] = *(const v2f*)(Arow + 4 * u);

  const int NSTAGE = HIDDEN / KSLICE;                // 128
  for (int s = 0; s < NSTAGE; ++s) {
    __syncthreads();                                 // prior stage reads retired
    *(float4*)(&Blds[row0 * BPAD + c0]) = st0;
    *(float4*)(&Blds[row1 * BPAD + c1]) = st1;
    __syncthreads();                                 // stage visible to all waves

    // prefetch stage s+1 (global loads overlap the 32 WMMAs below)
    if (s + 1 < NSTAGE) {
      const int kb = (s + 1) * KSLICE;
      st0 = *(const float4*)(Wst0 + kb);
      st1 = *(const float4*)(Wst1 + kb);
#pragma unroll
      for (int u = 0; u < 4; ++u) a_nxt[u] = *(const v2f*)(Arow + kb + 4 * u);
    }

    // ---- compute: 4 K-steps x 8 N-tiles from LDS ----
#pragma unroll
    for (int kk = 0; kk < 4; ++kk) {
      const int kc = 4 * kk + koff;
      v2f b[8];
#pragma unroll
      for (int t = 0; t < 8; ++t) {
        // B(K=kc..kc+1, N=n0+16t+r); padded stride -> conflict-free b64
        b[t] = *(const v2f*)(&Blds[(16 * t + r) * BPAD + kc]);
      }
#pragma unroll
      for (int t = 0; t < 8; ++t) acc[t] = wmma_f32(a_cur[kk], b[t], acc[t]);
    }
#pragma unroll
    for (int u = 0; u < 4; ++u) a_cur[u] = a_nxt[u];
  }

  if (mode == 1) {
#pragma unroll
    for (int t = 0; t < 8; ++t) {
      const int gn = n0 + 16 * t + r;
#pragma unroll
      for (int j = 0; j < 8; ++j) {
        const int gm   = m0 + j + 8 * hi;            // C/D layout: M = j + 8*hi
        const size_t i = (size_t)gm * HIDDEN + gn;
        out[i] = acc[t][j] + resid[i];
      }
    }
  } else {
#pragma unroll
    for (int t = 0; t < 8; ++t) {
      const int gn = n0 + 16 * t + r;
      const int h = gn >> 7, d = gn & (HEAD - 1);
#pragma unroll
      for (int j = 0; j < 8; ++j) {
        const int gm = m0 + j + 8 * hi;
        const int b = gm >> 11, sseq = gm & (SEQ - 1);
        out[((size_t)(b * HEADS + h) * SEQ + sseq) * HEAD + d] = acc[t][j];
      }
    }
  }
}

// ---------------------------------------------------------------------------
// RoPE, in place on k [B, H, S, D]. One thread per (b,h,s,i<64) pair.
// ---------------------------------------------------------------------------
__global__ void rope_kernel(float* __restrict__ k) {
  const int idx = blockIdx.x * blockDim.x + threadIdx.x;
  const int i  = idx & 63;
  const int s  = (idx >> 6) & (SEQ - 1);
  const int bh = idx >> 17;                         // b*HEADS + h
  // inv_freq = 10000^(-2i/128) = exp(-i * ln(10000)/64)
  const float inv = __expf((float)i * (-9.210340371976184f / 64.0f));
  const float ang = (float)s * inv;
  float sn, cs;
  sincosf(ang, &sn, &cs);                           // accurate arg reduction
  const size_t base = ((size_t)bh * SEQ + s) * HEAD;
  const float u1 = k[base + i];
  const float u2 = k[base + 64 + i];
  k[base + i]      = u1 * cs - u2 * sn;
  k[base + 64 + i] = u2 * cs + u1 * sn;
}

// ---------------------------------------------------------------------------
// TTT inner scan. One workgroup per (b,h); 8 waves; wave w owns W columns
// [16w, 16w+16) for the whole scan -> no intra-chunk barriers needed.
// Dynamic LDS: W 64KB | k_c 8KB | per-wave err 8KB  (80KB, OK on CDNA5 WGP).
// ---------------------------------------------------------------------------
__global__ __launch_bounds__(256) void ttt_scan(const float* __restrict__ kbuf,
                                                const float* __restrict__ vbuf,
                                                const float* __restrict__ W0,
                                                float* __restrict__ obuf) {
  extern __shared__ float smem[];
  float* Wl = smem;                                   // [128][128]
  float* kc = smem + HEAD * HEAD;                     // [16][128]
  float* errw = smem + HEAD * HEAD + MBS * HEAD + (threadIdx.x >> 5) * 256;

  const int tid  = threadIdx.x;
  const int lane = tid & 31;
  const int wave = tid >> 5;
  const int r    = lane & 15;
  const int hi   = lane >> 4;
  const int koff = 2 * hi;
  const int n0w  = wave * 16;                         // this wave's W columns
  const int bh   = blockIdx.x;

  // init W from W0[b,h]
  const float* W0p = W0 + (size_t)bh * HEAD * HEAD;
  for (int i = tid; i < HEAD * HEAD; i += 256) Wl[i] = W0p[i];

  const float* kbase = kbuf + (size_t)bh * SEQ * HEAD;
  const float* vbase = vbuf + (size_t)bh * SEQ * HEAD;
  const int bb = bh >> 4, hh = bh & 15;
  float* obase = obuf + (size_t)bb * SEQ * HIDDEN + hh * HEAD;

  const float gscale = 2.0f / (MBS * HEAD);           // 1/1024

  for (int c = 0; c < NCHUNK; ++c) {
    __syncthreads();                                  // kc reads of prev chunk done
    {
      const float* src = kbase + (size_t)c * MBS * HEAD;
      for (int i = tid; i < MBS * HEAD; i += 256) kc[i] = src[i];
    }
    __syncthreads();                                  // kc staged

    // ---- pred = k_c @ W (this wave's 16-wide N block) ----
    v8f p = vzero8();
    for (int kkb = 0; kkb < 32; kkb += 4) {
      v2f a[4], b[4];
#pragma unroll
      for (int u = 0; u < 4; ++u) {
        const int d0 = 4 * (kkb + u) + koff;
        a[u] = *(const v2f*)(kc + r * HEAD + d0);     // A(M=r, K=d0..d0+1)
        b[u].x = Wl[d0 * HEAD + n0w + r];             // B(K=d0,   N=n0w+r)
        b[u].y = Wl[(d0 + 1) * HEAD + n0w + r];       // B(K=d0+1, N)
      }
#pragma unroll
      for (int u = 0; u < 4; ++u) p = wmma_f32(a[u], b[u], p);
    }

    // ---- err = pred - v_c ; wave-private LDS stash (same-wave DS in-order) ----
    const float* vsrc = vbase + (size_t)c * MBS * HEAD + n0w + r;
#pragma unroll
    for (int j = 0; j < 8; ++j) {
      const int m = j + 8 * hi;                       // C/D layout row
      errw[m * 16 + r] = p[j] - vsrc[(size_t)m * HEAD];
    }

    // ---- grad = kc^T @ err ; W[:, block] -= (2/(MB*D)) * grad ----
#pragma unroll
    for (int mt = 0; mt < 8; ++mt) {
      v8f g = vzero8();
      v2f a[4], b[4];
#pragma unroll
      for (int kk = 0; kk < 4; ++kk) {
        const int m0 = 4 * kk + koff;
        a[kk].x = kc[m0 * HEAD + 16 * mt + r];        // A = kc^T
        a[kk].y = kc[(m0 + 1) * HEAD + 16 * mt + r];
        b[kk].x = errw[m0 * 16 + r];                  // B = err
        b[kk].y = errw[(m0 + 1) * 16 + r];
      }
#pragma unroll
      for (int kk = 0; kk < 4; ++kk) g = wmma_f32(a[kk], b[kk], g);
#pragma unroll
      for (int j = 0; j < 8; ++j) {
        const int d = 16 * mt + j + 8 * hi;
        Wl[d * HEAD + n0w + r] -= gscale * g[j];      // unique (d,e) per lane
      }
    }

    // ---- out = k_c @ W_new (reads only this wave's updated columns) ----
    v8f o = vzero8();
    for (int kkb = 0; kkb < 32; kkb += 4) {
      v2f a[4], b[4];
#pragma unroll
      for (int u = 0; u < 4; ++u) {
        const int d0 = 4 * (kkb + u) + koff;
        a[u] = *(const v2f*)(kc + r * HEAD + d0);
        b[u].x = Wl[d0 * HEAD + n0w + r];
        b[u].y = Wl[(d0 + 1) * HEAD + n0w + r];
      }
#pragma unroll
      for (int u = 0; u < 4; ++u) o = wmma_f32(a[u], b[u], o);
    }
#pragma unroll
    for (int j = 0; j < 8; ++j) {
      const int s = c * MBS + j + 8 * hi;
      obase[(size_t)s * HIDDEN + n0w + r] = o[j];
    }
  }
}

// ---------------------------------------------------------------------------
// Row LayerNorm (in place): one block per row of o [8192 x 2048].
// ---------------------------------------------------------------------------
__global__ __launch_bounds__(256) void ln_kernel(float* __restrict__ o,
                                                 const float* __restrict__ gam,
                                                 const float* __restrict__ bet) {
  __shared__ float rsum[256];
  __shared__ float rsq[256];
  const int tid = threadIdx.x;
  float* rp = o + (size_t)blockIdx.x * HIDDEN;

  float vals[8];
  {
    float4 a = *(const float4*)(rp + tid * 8);
    float4 b = *(const float4*)(rp + tid * 8 + 4);
    vals[0] = a.x; vals[1] = a.y; vals[2] = a.z; vals[3] = a.w;
    vals[4] = b.x; vals[5] = b.y; vals[6] = b.z; vals[7] = b.w;
  }
  float s = 0.f, sq = 0.f;
#pragma unroll
  for (int i = 0; i < 8; ++i) { s += vals[i]; sq += vals[i] * vals[i]; }
  rsum[tid] = s;
  rsq[tid]  = sq;
  __syncthreads();
  for (int off = 128; off >= 1; off >>= 1) {
    if (tid < off) {
      rsum[tid] += rsum[tid + off];
      rsq[tid]  += rsq[tid + off];
    }
    __syncthreads();
  }
  const float mu   = rsum[0] * (1.0f / HIDDEN);
  const float var  = rsq[0] * (1.0f / HIDDEN) - mu * mu;
  const float rstd = rsqrtf(var + 1e-5f);
  const int c0 = tid * 8;
#pragma unroll
  for (int i = 0; i < 8; ++i)
    rp[c0 + i] = (vals[i] - mu) * rstd * gam[c0 + i] + bet[c0 + i];
}

// ---------------------------------------------------------------------------
extern "C" void kernel_launch(void* const* d_in, const int* in_sizes, int n_in,
                              void* d_out, int out_size, void* d_ws,
                              size_t ws_size, hipStream_t stream) {
  (void)in_sizes; (void)n_in; (void)out_size; (void)ws_size;
  const float* x   = (const float*)d_in[0];
  const float* Wk  = (const float*)d_in[1];
  const float* Wv  = (const float*)d_in[2];
  const float* Wo  = (const float*)d_in[3];
  const float* lng = (const float*)d_in[4];
  const float* lnb = (const float*)d_in[5];
  const float* W0  = (const float*)d_in[6];
  float* out = (float*)d_out;

  float* ws   = (float*)d_ws;
  const size_t plane = (size_t)ROWS * HIDDEN;         // 16.78M floats = 64MB
  float* kbuf = ws;                                   // [B,H,S,D]
  float* vbuf = ws + plane;                           // [B,H,S,D]
  float* obuf = ws + 2 * plane;                       // [B,S,HIDDEN]

  dim3 gblk(ROWS / 128, HIDDEN / 128);                // 64 x 16 blocks

  gemm_xWT<<<gblk, 256, 0, stream>>>(x, Wk, kbuf, nullptr, 0);
  gemm_xWT<<<gblk, 256, 0, stream>>>(x, Wv, vbuf, nullptr, 0);

  const int nrope = BATCH * HEADS * SEQ * (HEAD / 2); // 8.39M threads
  rope_kernel<<<nrope / 256, 256, 0, stream>>>(kbuf);

  const size_t lds = (size_t)(HEAD * HEAD + MBS * HEAD + 8 * 256) * sizeof(float);
  ttt_scan<<<BATCH * HEADS, 256, lds, stream>>>(kbuf, vbuf, W0, obuf); // 80KB LDS

  ln_kernel<<<ROWS, 256, 0, stream>>>(obuf, lng, lnb);
  gemm_xWT<<<gblk, 256, 0, stream>>>(obuf, Wo, out, x, 1);
}